// Node2GraphAttention_61967788147266
// MI455X (gfx1250) — compile-verified
//
#include <hip/hip_runtime.h>

// Node2Graph attention pooling for MI455X (gfx1250, wave32).
// Bandwidth-bound: ~256 MB of n_embedding traffic dominates (ideal ~11 us at
// 23.3 TB/s). n_embedding is read from HBM once into LDS and reused for both
// the WMMA dot-product stage and the weighted segment accumulation. Sorted
// n_batch -> register accumulation per segment, rare f32 atomics on flush.

typedef __attribute__((ext_vector_type(2))) float v2f;
typedef __attribute__((ext_vector_type(4))) float v4f;
typedef __attribute__((ext_vector_type(8))) float v8f;

#define DIMS 128
#define LDS_STRIDE 132           // 128 + 4 pad: conflict-free strided reads
#define TILE 16                  // nodes per WMMA tile
#define TILES_PER_WAVE 8
#define NODES_PER_WAVE (TILE * TILES_PER_WAVE)   // 128
#define WAVES_PER_BLOCK 2
#define BLOCK (32 * WAVES_PER_BLOCK)

__global__ void n2g_zero_kernel(float* __restrict__ out, int n) {
    int i = blockIdx.x * blockDim.x + threadIdx.x;
    if (i < n) out[i] = 0.0f;
}

__global__ __launch_bounds__(BLOCK)
void n2g_attention_kernel(const float* __restrict__ nEmb,
                          const float* __restrict__ gEmb,
                          const int*   __restrict__ batch,
                          float*       __restrict__ out,
                          int N) {
    __shared__ float lds[WAVES_PER_BLOCK][2][TILE * LDS_STRIDE];
    const int lane  = threadIdx.x & 31;
    const int wave  = threadIdx.x >> 5;
    const int gwave = blockIdx.x * WAVES_PER_BLOCK + wave;
    float* nT = lds[wave][0];
    float* gT = lds[wave][1];

    const int m    = lane & 15;             // matrix row this lane serves
    const int off  = (lane < 16) ? 0 : 2;   // f32-WMMA K split across lane halves
    const int col4 = lane * 4;              // this lane's 4 output columns

    const long base0 = (long)gwave * NODES_PER_WAVE;

    int curG = -1;                          // current (sorted) segment id
    v4f acc  = {0.f, 0.f, 0.f, 0.f};

    for (int t = 0; t < TILES_PER_WAVE; ++t) {
        const long base = base0 + (long)t * TILE;

        // -------- per-node graph index, branchless (clamped) load --------
        long bi = base + m;
        if (bi > (long)N - 1) bi = (long)N - 1;
        const int myG = batch[bi];

        // -------- stage 16 node rows + 16 gathered graph rows in LDS -----
        #pragma unroll
        for (int r = 0; r < TILE; ++r) {
            const long node = base + r;                  // wave-uniform
            const int  gr   = __shfl(myG, r, 32);        // wave-uniform
            v4f nv = {0.f, 0.f, 0.f, 0.f};
            if (node < N)                                // uniform branch
                nv = *(const v4f*)(nEmb + node * DIMS + col4);
            *(v4f*)(nT + r * LDS_STRIDE + col4) = nv;
            v4f gv = *(const v4f*)(gEmb + (long)gr * DIMS + col4);
            *(v4f*)(gT + r * LDS_STRIDE + col4) = gv;
        }
        // cross-lane LDS RAW: make staged data visible before operand reads
        asm volatile("s_wait_dscnt 0" ::: "memory");

        // -------- 16x16 x K=128 cross products: 32 chained f32 WMMAs ----
        const float* pA = nT + m * LDS_STRIDE + off;
        const float* pB = gT + m * LDS_STRIDE + off;
        v8f c = {0.f, 0.f, 0.f, 0.f, 0.f, 0.f, 0.f, 0.f};
        #pragma unroll
        for (int k0 = 0; k0 < DIMS; k0 += 4) {
            v2f a = *(const v2f*)(pA + k0);
            v2f b = *(const v2f*)(pB + k0);
            c = __builtin_amdgcn_wmma_f32_16x16x4_f32(
                    false, a, false, b, (short)0, c, false, false);
        }

        // -------- diagonal -> per-node sigmoid coefficients (broadcast) --
        // D[m][m]: m<8 -> c[m] @ lane m ; m>=8 -> c[m-8] @ lane m+16
        float coef[TILE];
        #pragma unroll
        for (int j = 0; j < TILE; ++j) {
            const int srcLane = (j < 8) ? j : (j + 16);
            const float dj = __shfl(c[j & 7], srcLane, 32);
            coef[j] = 1.0f / (1.0f + __expf(-dj));
        }

        // -------- weighted accumulate, sorted-segment register flush -----
        #pragma unroll
        for (int j = 0; j < TILE; ++j) {
            const long node = base + j;                  // wave-uniform
            if (node < N) {
                const int gj = __shfl(myG, j, 32);       // wave-uniform value
                v4f row = *(const v4f*)(nT + j * LDS_STRIDE + col4);
                if (gj != curG) {                        // uniform branch
                    if (curG >= 0) {
                        float* o = out + (long)curG * DIMS + col4;
                        atomicAdd(o + 0, acc.x);
                        atomicAdd(o + 1, acc.y);
                        atomicAdd(o + 2, acc.z);
                        atomicAdd(o + 3, acc.w);
                    }
                    curG = gj;
                    acc = coef[j] * row;
                } else {
                    acc += coef[j] * row;
                }
            }
        }
    }

    if (curG >= 0) {
        float* o = out + (long)curG * DIMS + col4;
        atomicAdd(o + 0, acc.x);
        atomicAdd(o + 1, acc.y);
        atomicAdd(o + 2, acc.z);
        atomicAdd(o + 3, acc.w);
    }
}

extern "C" void kernel_launch(void* const* d_in, const int* in_sizes, int n_in,
                              void* d_out, int out_size, void* d_ws, size_t ws_size,
                              hipStream_t stream) {
    const float* nEmb  = (const float*)d_in[0];
    const float* gEmb  = (const float*)d_in[1];
    const int*   batch = (const int*)d_in[2];
    float*       out   = (float*)d_out;

    const int N = in_sizes[0] / DIMS;

    // zero the (poisoned) output accumulator
    n2g_zero_kernel<<<(out_size + 255) / 256, 256, 0, stream>>>(out, out_size);

    const int numWaves  = (N + NODES_PER_WAVE - 1) / NODES_PER_WAVE;
    const int numBlocks = (numWaves + WAVES_PER_BLOCK - 1) / WAVES_PER_BLOCK;
    n2g_attention_kernel<<<numBlocks, BLOCK, 0, stream>>>(nEmb, gEmb, batch, out, N);
}